// VectorQuantizer_17523466568354
// MI455X (gfx1250) — compile-verified
//
#include <hip/hip_runtime.h>

typedef float v2f __attribute__((ext_vector_type(2)));
typedef float v4f __attribute__((ext_vector_type(4)));
typedef float v8f __attribute__((ext_vector_type(8)));

#define N_TOT 65536
#define DDIM  64
#define KCB   1024

// ---------------------------------------------------------------------------
// Pass 1: C2h[k] = 0.5 * ||W[k]||^2   (argmin(d) == argmin(C2h - x.w))
// ---------------------------------------------------------------------------
__global__ __launch_bounds__(256) void vq_c2_kernel(const float* __restrict__ W,
                                                    float* __restrict__ C2h) {
    int k = blockIdx.x * blockDim.x + threadIdx.x;
    if (k < KCB) {
        const v4f* row = (const v4f*)(W + (size_t)k * DDIM);
        float s = 0.0f;
#pragma unroll
        for (int i = 0; i < DDIM / 4; ++i) {
            v4f v = row[i];
            s += v.x * v.x + v.y * v.y + v.z * v.z + v.w * v.w;
        }
        C2h[k] = 0.5f * s;
    }
}

// ---------------------------------------------------------------------------
// Pass 2: WMMA distance GEMM + running argmin + gather.
// Each wave owns 32 rows of x (two 16-row WMMA tiles, 2 independent
// accumulation chains) and sweeps all 64 codebook column tiles.
// ---------------------------------------------------------------------------
__global__ __launch_bounds__(256) void vq_main_kernel(const float* __restrict__ x,
                                                      const float* __restrict__ W,
                                                      const float* __restrict__ C2h,
                                                      float* __restrict__ outQ,
                                                      int*   __restrict__ outI) {
    const int lane     = threadIdx.x & 31;
    const int wave     = threadIdx.x >> 5;
    const int half     = lane >> 4;   // 0: lanes 0-15, 1: lanes 16-31
    const int ln       = lane & 15;
    const int rBase    = (blockIdx.x * 8 + wave) * 32;

    // ---- A fragments for both 16-row tiles, all 16 k-steps (register resident)
    // 16x4 f32 A layout: lane l -> row (l&15), cols (l>>4)*2 + {0,1}
    v2f xa0[16], xa1[16];
    {
        const float* xr0 = x + (size_t)(rBase + ln) * DDIM + half * 2;
        const float* xr1 = xr0 + 16 * DDIM;
#pragma unroll
        for (int s = 0; s < 16; ++s) {
            xa0[s] = *(const v2f*)(xr0 + 4 * s);
            xa1[s] = *(const v2f*)(xr1 + 4 * s);
        }
    }

    float bestV0[8], bestV1[8];
    int   bestI0[8], bestI1[8];
#pragma unroll
    for (int j = 0; j < 8; ++j) {
        bestV0[j] = 3.4e38f; bestV1[j] = 3.4e38f;
        bestI0[j] = 0;       bestI1[j] = 0;
    }

    for (int t = 0; t < 64; ++t) {
        const int c16 = t * 16;
        // B 4x16 layout mirrors A: lane l -> W row (c16 + (l&15)), cols (l>>4)*2+{0,1}
        const float* wr = W + (size_t)(c16 + ln) * DDIM + half * 2;
        if (t < 63) __builtin_prefetch(wr + 16 * DDIM, 0, 0);  // next tile -> global_prefetch

        v8f acc0 = {}, acc1 = {};
#pragma unroll
        for (int s = 0; s < 16; ++s) {
            v2f wb = *(const v2f*)(wr + 4 * s);
            acc0 = __builtin_amdgcn_wmma_f32_16x16x4_f32(
                false, xa0[s], false, wb, (short)0, acc0, false, false);
            acc1 = __builtin_amdgcn_wmma_f32_16x16x4_f32(
                false, xa1[s], false, wb, (short)0, acc1, false, false);
        }

        const float h = C2h[c16 + ln];      // 0.5*||w_n||^2 for this lane's column
        const int   n = c16 + ln;
#pragma unroll
        for (int j = 0; j < 8; ++j) {
            // D layout: acc[j] -> row M = j + 8*half, col N = ln
            float s0 = h - acc0[j];
            float s1 = h - acc1[j];
            if (s0 < bestV0[j]) { bestV0[j] = s0; bestI0[j] = n; }  // strict '<' keeps lowest n
            if (s1 < bestV1[j]) { bestV1[j] = s1; bestI1[j] = n; }
        }
    }

    // ---- argmin across the 16 lanes sharing each row (xor 1,2,4,8 stays in-half)
#pragma unroll
    for (int j = 0; j < 8; ++j) {
        float v0 = bestV0[j]; int i0 = bestI0[j];
        float v1 = bestV1[j]; int i1 = bestI1[j];
#pragma unroll
        for (int off = 1; off < 16; off <<= 1) {
            float ov0 = __shfl_xor(v0, off, 32); int oi0 = __shfl_xor(i0, off, 32);
            float ov1 = __shfl_xor(v1, off, 32); int oi1 = __shfl_xor(i1, off, 32);
            if (ov0 < v0 || (ov0 == v0 && oi0 < i0)) { v0 = ov0; i0 = oi0; }
            if (ov1 < v1 || (ov1 == v1 && oi1 < i1)) { v1 = ov1; i1 = oi1; }
        }
        bestI0[j] = i0; bestI1[j] = i1;
    }

    // ---- gather W[best] (16 lanes x float4 = 64 floats/row) + index write
#pragma unroll
    for (int j = 0; j < 8; ++j) {
        int row0 = rBase + j + 8 * half;
        int row1 = row0 + 16;
        int i0 = bestI0[j], i1 = bestI1[j];
        v4f q0 = *(const v4f*)(W + (size_t)i0 * DDIM + 4 * ln);
        v4f q1 = *(const v4f*)(W + (size_t)i1 * DDIM + 4 * ln);
        *(v4f*)(outQ + (size_t)row0 * DDIM + 4 * ln) = q0;
        *(v4f*)(outQ + (size_t)row1 * DDIM + 4 * ln) = q1;
        if (ln == 0) { outI[row0] = i0; outI[row1] = i1; }
    }
}

// ---------------------------------------------------------------------------
extern "C" void kernel_launch(void* const* d_in, const int* in_sizes, int n_in,
                              void* d_out, int out_size, void* d_ws, size_t ws_size,
                              hipStream_t stream) {
    const float* x = (const float*)d_in[0];   // [N, 64] fp32
    const float* W = (const float*)d_in[1];   // [K, 64] fp32
    float* C2h  = (float*)d_ws;               // K floats scratch
    float* outQ = (float*)d_out;              // quantized [N,64] fp32
    int*   outI = (int*)((float*)d_out + (size_t)N_TOT * DDIM);  // closest [N] int32

    vq_c2_kernel<<<KCB / 256, 256, 0, stream>>>(W, C2h);
    vq_main_kernel<<<N_TOT / 256, 256, 0, stream>>>(x, W, C2h, outQ, outI);
}